// SGCN_deform_s1_53403623358894
// MI455X (gfx1250) — compile-verified
//
#include <hip/hip_runtime.h>
#include <hip/hip_bf16.h>
#include <stdint.h>

typedef __attribute__((ext_vector_type(2))) float v2f;
typedef __attribute__((ext_vector_type(8))) float v8f;

#define HIDDIM 32

// Non-returning hardware fp32 atomic add (fire-and-forget, STOREcnt-tracked).
// Inline asm guarantees global_atomic_add_f32 on gfx1250 (no CAS-loop fallback).
__device__ __forceinline__ void atomic_add_f32(float* p, float v) {
  asm volatile("global_atomic_add_f32 %0, %1, off"
               :
               : "v"(p), "v"(v)
               : "memory");
}

// ---------------- elementwise helpers ----------------

__global__ void k_fill(float* __restrict__ p, float v, int n) {
  int i = blockIdx.x * blockDim.x + threadIdx.x;
  if (i < n) p[i] = v;
}

__global__ void k_scaled_copy(float* __restrict__ dst, const float* __restrict__ src,
                              float s, int n) {
  int i = blockIdx.x * blockDim.x + threadIdx.x;
  if (i < n) dst[i] = s * src[i];
}

// ---------------- graph normalization ----------------

__global__ void k_degree(const long long* __restrict__ ei, float* __restrict__ deg, int E) {
  int e = blockIdx.x * blockDim.x + threadIdx.x;
  if (e < E) atomic_add_f32(&deg[(int)ei[e]], 1.0f);
}

__global__ void k_dinv(float* __restrict__ deg, int n) {
  int i = blockIdx.x * blockDim.x + threadIdx.x;
  if (i < n) {
    float d = deg[i];
    deg[i] = (d > 0.0f) ? rsqrtf(fmaxf(d, 1.0f)) : 0.0f;
  }
}

__global__ void k_normw(const long long* __restrict__ ei, const float* __restrict__ dinv,
                        float* __restrict__ nw, int E) {
  int e = blockIdx.x * blockDim.x + threadIdx.x;
  if (e < E) {
    int s = (int)ei[e];
    int d = (int)ei[(long long)E + e];
    nw[e] = -dinv[s] * dinv[d];
  }
}

// ---------------- input layer: h = relu(x[N,3] @ W0[3,32] + b0) ----------------

__global__ void k_in_layer(const float* __restrict__ x, const float* __restrict__ W0,
                           const float* __restrict__ b0, float* __restrict__ h, int n) {
  int i = blockIdx.x * blockDim.x + threadIdx.x;
  if (i < n * HIDDIM) {
    int node = i >> 5, c = i & 31;
    float v = x[node * 3 + 0] * W0[0 * HIDDIM + c] +
              x[node * 3 + 1] * W0[1 * HIDDIM + c] +
              x[node * 3 + 2] * W0[2 * HIDDIM + c] + b0[c];
    h[i] = fmaxf(v, 0.0f);
  }
}

// ---------------- sparse propagation: y[dst] += nw[e] * x[src], 32 feats/edge ----
// 8 lanes per edge, float4 gather (128B per edge, L2-resident), 4 fp32 atomics/lane.

__global__ void k_prop(const long long* __restrict__ ei, const float* __restrict__ nw,
                       const float* __restrict__ xin, float* __restrict__ y, int E) {
  int t = blockIdx.x * blockDim.x + threadIdx.x;
  int e = t >> 3;
  int q = t & 7;
  if (e < E) {
    float w = nw[e];
    int s = (int)ei[e];
    int d = (int)ei[(long long)E + e];
    const float4 v = *(const float4*)(xin + (size_t)s * HIDDIM + q * 4);
    float* yd = y + (size_t)d * HIDDIM + q * 4;
    atomic_add_f32(yd + 0, w * v.x);
    atomic_add_f32(yd + 1, w * v.y);
    atomic_add_f32(yd + 2, w * v.z);
    atomic_add_f32(yd + 3, w * v.w);
  }
}

// ---------------- WMMA fp32 GEMM: OUT[16x16 tile] (+)= A[N,32] @ W[32,32] ----
// One wave per 16x16 tile; K=32 as 8 x v_wmma_f32_16x16x4_f32.
// Fused: optional accumulate, bias, residual add, relu on the final pass.

__launch_bounds__(128)
__global__ void k_gemm32_wmma(const float* __restrict__ A, const float* __restrict__ W,
                              float* __restrict__ OUT, const float* __restrict__ bias,
                              const float* __restrict__ res,
                              int accumulate, int do_relu, int n) {
  const int lane = threadIdx.x;                            // 0..31
  const int tile = blockIdx.x * blockDim.y + threadIdx.y;  // 16-row tile index
  const int m0 = tile * 16;
  if (m0 >= n) return;                                     // wave-uniform exit
  const int n0 = blockIdx.y * 16;
  const int ml = lane & 15;
  const bool hi = lane >= 16;
  const int kk = hi ? 2 : 0;                               // A/B K split across lane halves

  v8f c = {0, 0, 0, 0, 0, 0, 0, 0};
  if (accumulate) {
#pragma unroll
    for (int v = 0; v < 8; ++v) {
      int row = m0 + (hi ? 8 + v : v);
      if (row > n - 1) row = n - 1;
      c[v] = OUT[(size_t)row * HIDDIM + n0 + ml];
    }
  }

  int ra = m0 + ml;                                        // A row for this lane
  if (ra > n - 1) ra = n - 1;
  const float* arow = A + (size_t)ra * HIDDIM;

#pragma unroll
  for (int kb = 0; kb < HIDDIM; kb += 4) {
    v2f a, b;
    a.x = arow[kb + kk + 0];                               // A: M=ml, K=kb+kk / kb+kk+1
    a.y = arow[kb + kk + 1];
    b.x = W[(kb + kk + 0) * HIDDIM + n0 + ml];             // B: K rows striped across lanes
    b.y = W[(kb + kk + 1) * HIDDIM + n0 + ml];
    c = __builtin_amdgcn_wmma_f32_16x16x4_f32(false, a, false, b, (short)0, c,
                                              false, false);
  }

  const float bv = bias ? bias[n0 + ml] : 0.0f;
#pragma unroll
  for (int v = 0; v < 8; ++v) {
    int row = m0 + (hi ? 8 + v : v);
    if (row < n) {
      float val = c[v] + bv;
      if (res) val += res[(size_t)row * HIDDIM + n0 + ml];
      if (do_relu) val = fmaxf(val, 0.0f);
      OUT[(size_t)row * HIDDIM + n0 + ml] = val;
    }
  }
}

// ---------------- output layer: out[N] = h[N,32] @ W1[32,1] + b1 ----------------

__global__ void k_out_layer(const float* __restrict__ h, const float* __restrict__ W1,
                            const float* __restrict__ b1, float* __restrict__ out, int n) {
  int i = blockIdx.x * blockDim.x + threadIdx.x;
  if (i < n) {
    float acc = b1[0];
#pragma unroll
    for (int c = 0; c < HIDDIM; ++c) acc += h[(size_t)i * HIDDIM + c] * W1[c];
    out[i] = acc;
  }
}

// ---------------- host orchestration ----------------

extern "C" void kernel_launch(void* const* d_in, const int* in_sizes, int n_in,
                              void* d_out, int out_size, void* d_ws, size_t ws_size,
                              hipStream_t stream) {
  const float* x        = (const float*)d_in[0];
  const long long* ei   = (const long long*)d_in[1];   // int64 [2,E]: src then dst
  const float* W0       = (const float*)d_in[2];
  const float* b0       = (const float*)d_in[3];
  const float* c11W     = (const float*)d_in[4];
  const float* c11b     = (const float*)d_in[5];
  const float* c12W     = (const float*)d_in[6];
  const float* c12b     = (const float*)d_in[7];
  const float* c21W     = (const float*)d_in[8];
  const float* c21b     = (const float*)d_in[9];
  const float* c22W     = (const float*)d_in[10];
  const float* c22b     = (const float*)d_in[11];
  const float* W1       = (const float*)d_in[12];
  const float* b1       = (const float*)d_in[13];

  const int N  = in_sizes[0] / 3;
  const int E  = in_sizes[1] / 2;
  const int NF = N * HIDDIM;
  const int WSZ = HIDDIM * HIDDIM;

  // carve workspace (~142 MB): deg[N], nw[E], W2 scratch, 5 x feature buffers [N,32]
  float* ws = (float*)d_ws;
  size_t off = 0;
  auto carve = [&](size_t nflt) { float* p = ws + off; off += (nflt + 3) & ~(size_t)3; return p; };
  float* deg = carve(N);
  float* nw  = carve(E);
  float* W2s = carve(WSZ);   // 2*W[2], pre-scaled once per conv
  float* H   = carve(NF);
  float* Ab  = carve(NF);
  float* Bb  = carve(NF);
  float* T1  = carve(NF);
  float* P   = carve(NF);
  (void)ws_size; (void)n_in; (void)out_size;

  const int TB = 256;
  auto blks = [&](long long t) { return (int)((t + TB - 1) / TB); };

  // edge normalization weights (deg reused in-place as d^{-1/2})
  k_fill  <<<blks(N), TB, 0, stream>>>(deg, 0.0f, N);
  k_degree<<<blks(E), TB, 0, stream>>>(ei, deg, E);
  k_dinv  <<<blks(N), TB, 0, stream>>>(deg, N);
  k_normw <<<blks(E), TB, 0, stream>>>(ei, deg, nw, E);

  // input layer
  k_in_layer<<<blks(NF), TB, 0, stream>>>(x, W0, b0, H, N);

  const int tiles = (N + 15) / 16;
  dim3 ggrid((tiles + 3) / 4, 2, 1);   // y: two 16-col halves of HID=32
  dim3 gblk(32, 4, 1);                 // 4 waves per block, one tile each

  auto gemm = [&](const float* Ain, const float* Wc, float* OUT, const float* bias,
                  const float* res, int acc, int relu) {
    k_gemm32_wmma<<<ggrid, gblk, 0, stream>>>(Ain, Wc, OUT, bias, res, acc, relu, N);
  };

  // ChebConv K=3: OUT = T0@W0 + T1@W1 + T2@W2 + b, T1=prop(T0), T2=2*prop(T1)-T0
  // T2 trick: P = -0.5*T0 + prop(T1); final pass uses W2s = 2*W[2] so OUT += P@W2s.
  auto cheb = [&](const float* Xin, const float* Wc, const float* bc,
                  float* OUT, const float* res, int relu) {
    gemm(Xin, Wc + 0 * WSZ, OUT, nullptr, nullptr, 0, 0);
    k_fill<<<blks(NF), TB, 0, stream>>>(T1, 0.0f, NF);
    k_prop<<<blks((long long)E * 8), TB, 0, stream>>>(ei, nw, Xin, T1, E);
    gemm(T1, Wc + 1 * WSZ, OUT, nullptr, nullptr, 1, 0);
    k_scaled_copy<<<blks(NF), TB, 0, stream>>>(P, Xin, -0.5f, NF);
    k_scaled_copy<<<blks(WSZ), TB, 0, stream>>>(W2s, Wc + 2 * WSZ, 2.0f, WSZ);
    k_prop<<<blks((long long)E * 8), TB, 0, stream>>>(ei, nw, T1, P, E);
    gemm(P, W2s, OUT, bc, res, 1, relu);
  };

  // block 1: A = relu(cheb(H)); B = relu(cheb(A) + H)
  cheb(H,  c11W, c11b, Ab, nullptr, 1);
  cheb(Ab, c12W, c12b, Bb, H, 1);
  // block 2: A = relu(cheb(B)); H = relu(cheb(A) + B)
  cheb(Bb, c21W, c21b, Ab, nullptr, 1);
  cheb(Ab, c22W, c22b, H, Bb, 1);

  // output layer
  k_out_layer<<<blks(N), TB, 0, stream>>>(H, W1, b1, (float*)d_out, N);
}